// GCN_26182120636970
// MI455X (gfx1250) — compile-verified
//
#include <hip/hip_runtime.h>
#include <hip/hip_bf16.h>
#include <math.h>

typedef __attribute__((ext_vector_type(2))) float v2f;
typedef __attribute__((ext_vector_type(8))) float v8f;

// ---------------------------------------------------------------------------
// Degree accumulation: deg[dst] += 1 per edge (self-loop added in rsqrt pass)
// ---------------------------------------------------------------------------
__global__ void gcn_degree_kernel(const int* __restrict__ dst,
                                  float* __restrict__ deg, int nEdges) {
    int e = blockIdx.x * blockDim.x + threadIdx.x;
    if (e < nEdges) atomicAdd(&deg[dst[e]], 1.0f);
}

__global__ void gcn_rsqrt_kernel(float* __restrict__ dinv, int n) {
    int i = blockIdx.x * blockDim.x + threadIdx.x;
    if (i < n) dinv[i] = rsqrtf(dinv[i] + 1.0f);  // +1 = self loop; deg >= 1
}

// ---------------------------------------------------------------------------
// FP32 WMMA GEMM: H[N,FOUT] = X[N,128] @ W[128,FOUT]
// One wave computes a 16-row tile across the full padded width FPAD.
//
// W staged in LDS K-pair-interleaved: element (k,n) at
//   (k>>1)*PSTRIDE + 2*n + (k&1)
// so each B fragment (k,k+1 at column n) is a single contiguous, 8B-aligned
// ds_load_b64 straight into an even VGPR pair (no v_mov repacking).
// PSTRIDE chosen so PSTRIDE % 64 == 32: lanes 16-31 (khalf=2 -> +1 pair row)
// hit the disjoint half of the 64 LDS banks -> conflict-free b64 loads.
//
// A 16x4 f32 frag layout (ISA 7.12.2): lane L -> row M=L&15, K pair
// (L>>4)*2 in v[0], +1 in v[1]. B 4x16 symmetric: lane L -> col N=L&15.
// D: VGPR r -> M = r + (L>=16)*8, N = L&15.
// ---------------------------------------------------------------------------
template <int FOUT, int FPAD>
__global__ __launch_bounds__(256)
void gcn_gemm_wmma_kernel(const float* __restrict__ X,
                          const float* __restrict__ W,
                          float* __restrict__ H, int nTiles) {
    constexpr int FIN    = 128;
    constexpr int NT     = FPAD / 16;   // output column tiles
    constexpr int PSTRIDE = ((2 * FPAD) % 64 == 32) ? (2 * FPAD) : (2 * FPAD + 32);
    __shared__ float Wl[(FIN / 2) * PSTRIDE];

    for (int i = threadIdx.x; i < FIN * FPAD; i += blockDim.x) {
        int k = i / FPAD, n = i % FPAD;
        Wl[(k >> 1) * PSTRIDE + 2 * n + (k & 1)] =
            (n < FOUT) ? W[k * FOUT + n] : 0.0f;
    }
    __syncthreads();

    const int wave = threadIdx.x >> 5;
    const int lane = threadIdx.x & 31;
    const int tile = blockIdx.x * 8 + wave;
    if (tile >= nTiles) return;               // uniform per wave: EXEC all-1s

    const int m     = lane & 15;              // row within tile (A) / col (B,D)
    const int khalf = (lane >> 4) * 2;        // K sub-offset: 0 or 2

    v8f acc[NT] = {};
    const float* xrow  = X + (size_t)(tile * 16 + m) * FIN;
    const float* wbase = Wl + (lane >> 4) * PSTRIDE + 2 * m;

#pragma unroll 4
    for (int k0 = 0; k0 < FIN; k0 += 4) {
        const v2f a = *(const v2f*)(xrow + k0 + khalf);        // global_load_b64
        const float* wk = wbase + (k0 >> 1) * PSTRIDE;
#pragma unroll
        for (int t = 0; t < NT; ++t) {
            const v2f b = *(const v2f*)(wk + 32 * t);          // ds_load_b64
            acc[t] = __builtin_amdgcn_wmma_f32_16x16x4_f32(
                false, a, false, b, (short)0, acc[t], false, false);
        }
    }

    const int mOff = (lane >> 4) * 8;
#pragma unroll
    for (int t = 0; t < NT; ++t) {
        const int col = t * 16 + m;
        if (FOUT == FPAD || col < FOUT) {
            float* hrow = H + (size_t)(tile * 16 + mOff) * FOUT + col;
#pragma unroll
            for (int r = 0; r < 8; ++r)
                hrow[(size_t)r * FOUT] = acc[t][r];
        }
    }
}

// ---------------------------------------------------------------------------
// Edge aggregation: one wave per (edge | self-loop) task.
// out[dst] += h[src] * dinv[src]*dinv[dst]  (f32 atomics, L2-resident)
// Lanes cover channels c, c+32, ... -> coalesced loads & atomics.
// ---------------------------------------------------------------------------
template <int F>
__global__ __launch_bounds__(256)
void gcn_aggregate_kernel(const float* __restrict__ H,
                          float* __restrict__ Out,
                          const int* __restrict__ src,
                          const int* __restrict__ dst,
                          const float* __restrict__ dinv,
                          int nEdges, int nNodes) {
    const int task = blockIdx.x * (blockDim.x >> 5) + (threadIdx.x >> 5);
    const int lane = threadIdx.x & 31;
    if (task >= nEdges + nNodes) return;

    int s, d;
    if (task < nEdges) { s = src[task]; d = dst[task]; }
    else               { s = d = task - nEdges; }

    const float norm = dinv[s] * dinv[d];
    const float* hs = H + (size_t)s * F;
    float* od = Out + (size_t)d * F;
#pragma unroll
    for (int c = lane; c < F; c += 32)
        atomicAdd(&od[c], hs[c] * norm);
}

// ---------------------------------------------------------------------------
// Elementwise bias + ReLU (layers 1 & 2), in place.
// ---------------------------------------------------------------------------
template <int F>
__global__ void gcn_bias_relu_kernel(float* __restrict__ A,
                                     const float* __restrict__ b, long total) {
    long i = (long)blockIdx.x * blockDim.x + threadIdx.x;
    if (i < total) {
        float v = A[i] + b[(int)(i % F)];
        A[i] = fmaxf(v, 0.0f);
    }
}

// ---------------------------------------------------------------------------
// Final: out = log_softmax(agg + b3), wave per node, F = 40.
// ---------------------------------------------------------------------------
__global__ __launch_bounds__(256)
void gcn_logsoftmax_kernel(const float* __restrict__ A,
                           const float* __restrict__ b,
                           float* __restrict__ out, int nNodes) {
    constexpr int F = 40;
    const int node = blockIdx.x * (blockDim.x >> 5) + (threadIdx.x >> 5);
    const int lane = threadIdx.x & 31;
    if (node >= nNodes) return;

    const float* a = A + (size_t)node * F;
    float v0 = (lane < F)      ? a[lane]      + b[lane]      : -INFINITY;
    float v1 = (lane + 32 < F) ? a[lane + 32] + b[lane + 32] : -INFINITY;

    float m = fmaxf(v0, v1);
#pragma unroll
    for (int off = 16; off > 0; off >>= 1)
        m = fmaxf(m, __shfl_xor(m, off, 32));

    float e = 0.0f;
    if (lane < F)      e += expf(v0 - m);
    if (lane + 32 < F) e += expf(v1 - m);
#pragma unroll
    for (int off = 16; off > 0; off >>= 1)
        e += __shfl_xor(e, off, 32);

    const float lse = m + logf(e);
    float* o = out + (size_t)node * F;
    if (lane < F)      o[lane]      = v0 - lse;
    if (lane + 32 < F) o[lane + 32] = v1 - lse;
}

// ---------------------------------------------------------------------------
extern "C" void kernel_launch(void* const* d_in, const int* in_sizes, int n_in,
                              void* d_out, int out_size, void* d_ws, size_t ws_size,
                              hipStream_t stream) {
    const float* x  = (const float*)d_in[0];
    const int*   ei = (const int*)d_in[1];
    const float* W1 = (const float*)d_in[2];
    const float* b1 = (const float*)d_in[3];
    const float* W2 = (const float*)d_in[4];
    const float* b2 = (const float*)d_in[5];
    const float* W3 = (const float*)d_in[6];
    const float* b3 = (const float*)d_in[7];

    const int FIN = 128, FHID = 128, FOUTC = 40;
    const int N = in_sizes[0] / FIN;        // 50000
    const int E = in_sizes[1] / 2;          // 800000
    const int* src = ei;
    const int* dst = ei + E;

    // workspace layout (floats): dinv[N] | h[N*128] | agg[N*128]
    float* dinv = (float*)d_ws;
    size_t off  = ((size_t)N + 63) & ~(size_t)63;
    float* h    = dinv + off;
    float* agg  = h + (size_t)N * FHID;

    const int nTiles     = N / 16;                     // 50000 % 16 == 0
    const int gemmBlocks = (nTiles + 7) / 8;
    const int tasks      = E + N;
    const int aggBlocks  = (tasks + 7) / 8;            // 8 waves / block

    // degrees -> dinv
    hipMemsetAsync(dinv, 0, (size_t)N * sizeof(float), stream);
    gcn_degree_kernel<<<(E + 255) / 256, 256, 0, stream>>>(dst, dinv, E);
    gcn_rsqrt_kernel<<<(N + 255) / 256, 256, 0, stream>>>(dinv, N);

    // ---- layer 1 ----
    gcn_gemm_wmma_kernel<128, 128><<<gemmBlocks, 256, 0, stream>>>(x, W1, h, nTiles);
    hipMemsetAsync(agg, 0, (size_t)N * FHID * sizeof(float), stream);
    gcn_aggregate_kernel<128><<<aggBlocks, 256, 0, stream>>>(h, agg, src, dst, dinv, E, N);
    {
        long total = (long)N * FHID;
        gcn_bias_relu_kernel<128><<<(int)((total + 255) / 256), 256, 0, stream>>>(agg, b1, total);
    }

    // ---- layer 2 ----
    gcn_gemm_wmma_kernel<128, 128><<<gemmBlocks, 256, 0, stream>>>(agg, W2, h, nTiles);
    hipMemsetAsync(agg, 0, (size_t)N * FHID * sizeof(float), stream);
    gcn_aggregate_kernel<128><<<aggBlocks, 256, 0, stream>>>(h, agg, src, dst, dinv, E, N);
    {
        long total = (long)N * FHID;
        gcn_bias_relu_kernel<128><<<(int)((total + 255) / 256), 256, 0, stream>>>(agg, b2, total);
    }

    // ---- layer 3 (FOUT=40, padded to 48 cols for WMMA tiles) ----
    gcn_gemm_wmma_kernel<40, 48><<<gemmBlocks, 256, 0, stream>>>(agg, W3, h, nTiles);
    hipMemsetAsync(agg, 0, (size_t)N * FOUTC * sizeof(float), stream);
    gcn_aggregate_kernel<40><<<aggBlocks, 256, 0, stream>>>(h, agg, src, dst, dinv, E, N);

    gcn_logsoftmax_kernel<<<(N + 7) / 8, 256, 0, stream>>>(agg, b3, (float*)d_out, N);
}